// RoiPoolingConv_10952166605115
// MI455X (gfx1250) — compile-verified
//
#include <hip/hip_runtime.h>

#define POOL 7
#define NUM_ROIS 300
#define IMG_H 200
#define IMG_W 200
#define CHANNELS 512

typedef __attribute__((ext_vector_type(2))) float v2f;
typedef __attribute__((ext_vector_type(8))) float v8f;

// One block (8 wave32) per (roi, output-row py).
// GEMM view: out[px, c] = sum_{k=0..27} A[px,k] * B[k,c]
//   k = a*14 + spx*2 + b ; source vector = img[Y_a(py), X_b(spx), :]
//   A[px,k] = (spx==px) ? wy_a * wx_b(px) : 0   (4 nonzeros per row)
// Evaluated as 7 chained V_WMMA_F32_16X16X4_F32 per 16-channel group.
__global__ __launch_bounds__(256) void roi_align_wmma(
    const float* __restrict__ img, const int* __restrict__ rois,
    float* __restrict__ out)
{
  const int lane = threadIdx.x & 31;
  const int wave = threadIdx.x >> 5;
  const int blk  = blockIdx.x;            // 0 .. NUM_ROIS*POOL-1
  const int r    = blk / POOL;
  const int py   = blk - r * POOL;

  const int rx = rois[r * 4 + 0];
  const int ry = rois[r * 4 + 1];
  const int rw = rois[r * 4 + 2];
  const int rh = rois[r * 4 + 3];

  const float hstep = (float)rh / (float)POOL;
  const float wstep = (float)rw / (float)POOL;

  // y interpolation for this output row (matches TF bilinear, align_corners=False)
  const float sy = (float)py * hstep;
  const int   y0 = (int)floorf(sy);
  const float fy = sy - (float)y0;
  const int   y1 = min(y0 + 1, rh - 1);
  const int   Y0 = min(max(ry + y0, 0), IMG_H - 1);
  const int   Y1 = min(max(ry + y1, 0), IMG_H - 1);

  const int m  = lane & 15;               // A-matrix row (output px when < 7)
  const int hi = (lane >> 4) & 1;         // upper half-wave -> K offset +2

  // x weights for this lane's output pixel
  float wx0m = 0.f, wx1m = 0.f;
  if (m < POOL) {
    const float sx = (float)m * wstep;
    const int   x0 = (int)floorf(sx);
    const float fx = sx - (float)x0;
    wx0m = 1.0f - fx;
    wx1m = fx;
  }
  const float wy0 = 1.0f - fy;
  const float wy1 = fy;

  // Per-lane A fragment values and B gather offsets for each K-slice kk.
  float a0[POOL], a1[POOL];
  int   boff0[POOL], boff1[POOL];

#pragma unroll
  for (int kk = 0; kk < POOL; ++kk) {
    const int kbase = kk * 4 + hi * 2;
#pragma unroll
    for (int j = 0; j < 2; ++j) {
      const int k   = kbase + j;          // 0..27
      const int a   = (k >= 14) ? 1 : 0;
      const int rem = k - 14 * a;
      const int spx = rem >> 1;
      const int b   = rem & 1;
      // A value (ISA 16x4 layout: VGPR0 = K {0,2}, VGPR1 = K {1,3} per half-wave)
      float av = 0.0f;
      if (m < POOL && spx == m)
        av = (a ? wy1 : wy0) * (b ? wx1m : wx0m);
      // B source element offset for this lane (lane&15 = channel-in-group)
      const float sxs = (float)spx * wstep;
      const int   x0s = (int)floorf(sxs);
      const int   x1s = min(x0s + 1, rw - 1);
      const int   Xs  = min(max(rx + (b ? x1s : x0s), 0), IMG_W - 1);
      const int   Ys  = a ? Y1 : Y0;
      const int   off = (Ys * IMG_W + Xs) * CHANNELS + (lane & 15);
      if (j == 0) { a0[kk] = av; boff0[kk] = off; }
      else        { a1[kk] = av; boff1[kk] = off; }
    }
  }

  // Each wave owns 4 channel groups of 16 (8 waves cover all 512 channels).
  const size_t orow = ((size_t)(r * POOL + py) * POOL) * CHANNELS;

#pragma unroll
  for (int i = 0; i < 4; ++i) {
    const int cg = wave * 4 + i;
    const int co = cg * 16;
    v8f acc = {0.f, 0.f, 0.f, 0.f, 0.f, 0.f, 0.f, 0.f};
#pragma unroll
    for (int kk = 0; kk < POOL; ++kk) {
      v2f bfr;
      bfr.x = img[boff0[kk] + co];
      bfr.y = img[boff1[kk] + co];
      v2f afr;
      afr.x = a0[kk];
      afr.y = a1[kk];
      // D = A(16x4) * B(4x16) + C  in f32 (exact, RNE)
      acc = __builtin_amdgcn_wmma_f32_16x16x4_f32(
          false, afr, false, bfr, (short)0, acc, false, false);
    }
    // D layout: VGPR j = row M=j (lanes 0-15) / M=j+8 (lanes 16-31), N = lane&15.
    // Valid rows are M=0..6 -> lanes 0-15, VGPRs 0..6.
    if (lane < 16) {
      const size_t ob = orow + (size_t)co + (size_t)lane;
#pragma unroll
      for (int j = 0; j < POOL; ++j)
        out[ob + (size_t)j * CHANNELS] = acc[j];
    }
  }
}

extern "C" void kernel_launch(void* const* d_in, const int* in_sizes, int n_in,
                              void* d_out, int out_size, void* d_ws, size_t ws_size,
                              hipStream_t stream) {
  const float* img  = (const float*)d_in[0];   // (1, 200, 200, 512) f32
  const int*   rois = (const int*)d_in[1];     // (1, 300, 4) i32: x,y,w,h
  float* out = (float*)d_out;                  // (1, 300, 7, 7, 512) f32
  (void)in_sizes; (void)n_in; (void)out_size; (void)d_ws; (void)ws_size;

  dim3 grid(NUM_ROIS * POOL);
  dim3 block(256);
  hipLaunchKernelGGL(roi_align_wmma, grid, block, 0, stream, img, rois, out);
}